// STMDecoder_42391327211935
// MI455X (gfx1250) — compile-verified
//
#include <hip/hip_runtime.h>
#include <hip/hip_bf16.h>

typedef __attribute__((ext_vector_type(2))) float v2f;
typedef __attribute__((ext_vector_type(8))) float v8f;
typedef __attribute__((ext_vector_type(4))) unsigned int u32x4;
typedef __attribute__((ext_vector_type(8))) int i32x8;
typedef __attribute__((ext_vector_type(4))) int i32x4;

#if defined(__has_builtin)
#if __has_builtin(__builtin_amdgcn_tensor_load_to_lds)
#define USE_TDM 1
#endif
#endif
#ifndef USE_TDM
#define USE_TDM 0
#endif

__device__ __forceinline__ v8f wmma4(v2f a, v2f b, v8f c) {
  // D = A(16x4,f32) * B(4x16,f32) + C(16x16,f32)
  return __builtin_amdgcn_wmma_f32_16x16x4_f32(
      /*neg_a=*/false, a, /*neg_b=*/false, b,
      /*c_mod=*/(short)0, c, /*reuse_a=*/false, /*reuse_b=*/false);
}

#if USE_TDM
// TDM 2D tile load: tileRows x tileCols f32 tile of a row-major matrix
// (row pitch = pitch elements) -> LDS at byte offset lds_off, row-major,
// optionally LDS-padded via g1flags (pad_enable/interval/amount fields).
// OOB rows (>= tensorRows) are zero-filled by hardware. Tracked by TENSORcnt.
__device__ __forceinline__ void tdm_load_tile_f32(
    unsigned int lds_off, const float* gaddr, unsigned int pitch,
    unsigned int tensorRows, unsigned int tileCols, unsigned int tileRows,
    unsigned int g1flags) {
  const unsigned long long ga = (unsigned long long)(uintptr_t)gaddr;
  u32x4 g0;
  g0[0] = 1u;                                   // count=1, user D#, no gather
  g0[1] = lds_off;                              // D#.lds_addr  (bits 63:32)
  g0[2] = (unsigned int)(ga & 0xFFFFFFFFu);     // global_addr  (bits 95:64)
  g0[3] = (unsigned int)((ga >> 32) & 0x01FFFFFFu)  // global_addr bits 56:32
        | (2u << 30);                           // D#.type = 2 ("image")
  i32x8 g1;
  g1[0] = (int)((2u << 16) | g1flags);          // data_size=2 (4B), wg_mask=0
  g1[1] = (int)((pitch & 0xFFFFu) << 16);       // tensor_dim0 low16
  g1[2] = (int)((pitch >> 16) | ((tensorRows & 0xFFFFu) << 16));
  g1[3] = (int)((tensorRows >> 16) | (tileCols << 16));   // tile_dim0
  g1[4] = (int)(tileRows & 0xFFFFu);            // tile_dim1 (tile_dim2 = 0)
  g1[5] = (int)pitch;                           // tensor_dim0_stride low32
  g1[6] = 0;                                    // stride0 hi / stride1 lo
  g1[7] = 0;                                    // stride1 hi
  const i32x4 z4 = {0, 0, 0, 0};                // groups 2/3 unused (2D)
  const i32x8 z8 = {0, 0, 0, 0, 0, 0, 0, 0};    // extended group unused
  __builtin_amdgcn_tensor_load_to_lds(g0, g1, z4, z4, z8, 0);
}
// pad: insert 4 DWORDs after every 64 DWORDs -> LDS row pitch 68 (bank-safe)
#define TDM_PAD_64_TO_68 ((1u << 20) | (5u << 22) | (3u << 25))
#endif

// ---------------------------------------------------------------------------
// LDS-tiled, double-buffered GEMM:
//   C[M,N] = A[M,K_total] @ B[K,N] (+bias) (+res), A row stride = lda.
// Block = 256 threads = 8 waves (2x4), block tile = 32x128, K chunk = 64.
// A and B tiles streamed by the Tensor Data Mover (wave 0 issues; A uses the
// D# LDS-pad feature for a conflict-free 68-DWORD pitch; TENSORcnt sync).
// grid.y = split-K segment. Requires K % 64 == 0, N % 128 == 0.
// ---------------------------------------------------------------------------
#define MBLK 32
#define NBLK 128
#define KC   64
#define SA_P 68   // padded A pitch: lane lr hits bank (lr*4 + c) -> conflict-free

__global__ __launch_bounds__(256) void gemm16_kernel(
    const float* __restrict__ A, int lda,
    const float* __restrict__ B,
    const float* __restrict__ bias, const float* __restrict__ res,
    float* __restrict__ C, int M, int N, int K,
    int kStep, unsigned long long cStride) {
  __shared__ float sA[2][MBLK * SA_P];
  __shared__ float sB[2][KC * NBLK];

  const int tid  = threadIdx.x;
  const int lane = tid & 31;
  const int wave = tid >> 5;
  const int half = lane >> 4;
  const int lr   = lane & 15;
  const int wm   = wave >> 2;   // 0..1 : 16-row slice
  const int wn   = wave & 3;    // 0..3 : 32-col slice

  const int ntN = N / NBLK;
  const int bm  = blockIdx.x / ntN;
  const int bn  = blockIdx.x - bm * ntN;
  const int m0  = bm * MBLK;
  const int n0  = bn * NBLK;

  // split-K offsets
  A += (size_t)blockIdx.y * kStep;
  B += (size_t)blockIdx.y * kStep * N;
  C += (size_t)blockIdx.y * cStride;

#if USE_TDM
  auto issueAB = [&](int ck, int buf) {    // wave 0 only: two tile DMAs
    // A: 32x64 tile, row pitch lda; LDS pitch padded 64->68; M-edge rows
    // zero-filled by tensor_dim1 = M - m0.
    tdm_load_tile_f32((unsigned int)(uintptr_t)&sA[buf][0],
                      A + (size_t)m0 * lda + ck * KC,
                      (unsigned int)lda, (unsigned int)(M - m0),
                      KC, MBLK, TDM_PAD_64_TO_68);
    // B: 64x128 tile, row pitch N, packed row-major.
    tdm_load_tile_f32((unsigned int)(uintptr_t)&sB[buf][0],
                      B + (size_t)(ck * KC) * N + n0,
                      (unsigned int)N, (unsigned int)K, NBLK, KC, 0u);
  };
#else
  float4 ra[2], rb[8];
  auto fetchAB = [&](int ck) {
    const int k0 = ck * KC;
#pragma unroll
    for (int j = 0; j < 2; ++j) {          // A: 32x64 = 512 float4
      const int e   = tid + j * 256;
      const int row = e >> 4, c4 = e & 15;
      int gr = m0 + row;
      gr = gr < M ? gr : M - 1;            // clamp; garbage rows never stored
      ra[j] = *(const float4*)(A + (size_t)gr * lda + k0 + c4 * 4);
    }
#pragma unroll
    for (int j = 0; j < 8; ++j) {          // B: 64x128 = 2048 float4
      const int e   = tid + j * 256;
      const int row = e >> 5, c4 = e & 31;
      rb[j] = *(const float4*)(B + (size_t)(k0 + row) * N + n0 + c4 * 4);
    }
  };
  auto stageAB = [&](int buf) {
#pragma unroll
    for (int j = 0; j < 2; ++j) {
      const int e   = tid + j * 256;
      const int row = e >> 4, c4 = e & 15;
      *(float4*)&sA[buf][row * SA_P + c4 * 4] = ra[j];
    }
#pragma unroll
    for (int j = 0; j < 8; ++j) {
      const int e = tid + j * 256;
      *(float4*)&sB[buf][e * 4] = rb[j];
    }
  };
#endif

  v8f acc[2] = {};
  auto compute = [&](int buf) {
#pragma unroll
    for (int k = 0; k < KC; k += 4) {
      const int ka = k + 2 * half;                 // ISA A/B operand K mapping
      v2f a = *(const v2f*)&sA[buf][(wm * 16 + lr) * SA_P + ka];
#pragma unroll
      for (int i = 0; i < 2; ++i) {
        v2f b;
        b.x = sB[buf][ka * NBLK + wn * 32 + i * 16 + lr];
        b.y = sB[buf][(ka + 1) * NBLK + wn * 32 + i * 16 + lr];
        acc[i] = wmma4(a, b, acc[i]);
      }
    }
  };

  const int nChunks = K / KC;
#if USE_TDM
  if (wave == 0) {
    issueAB(0, 0);
    __builtin_amdgcn_s_wait_tensorcnt(0);
  }
  __syncthreads();
  for (int c = 0; c < nChunks; ++c) {
    const int buf = c & 1;
    if (c + 1 < nChunks && wave == 0)
      issueAB(c + 1, buf ^ 1);            // TDM streams next tiles...
    compute(buf);                         // ...behind 32 WMMAs
    if (c + 1 < nChunks && wave == 0)
      __builtin_amdgcn_s_wait_tensorcnt(0);
    __syncthreads();
  }
#else
  fetchAB(0);
  stageAB(0);
  __syncthreads();
  for (int c = 0; c < nChunks; ++c) {
    const int buf = c & 1;
    if (c + 1 < nChunks) fetchAB(c + 1);
    compute(buf);
    if (c + 1 < nChunks) stageAB(buf ^ 1);
    __syncthreads();
  }
#endif

  const int colBase = n0 + wn * 32;
#pragma unroll
  for (int i = 0; i < 2; ++i) {
    const int col = colBase + i * 16 + lr;
    const float bs = bias ? bias[col] : 0.0f;
#pragma unroll
    for (int r = 0; r < 8; ++r) {
      const int m = m0 + wm * 16 + r + 8 * half;   // ISA D-layout row
      if (m < M) {
        float v = acc[i][r] + bs;
        if (res) v += res[(size_t)m * N + col];
        C[(size_t)m * N + col] = v;
      }
    }
  }
}

// ---------------------------------------------------------------------------
// Adaptive sampling: per (b,n,g,p) point, bilinear gather of 64 channels
// from 4 pyramid levels blended with softmax level weights.
// Block = 256 threads = 4 points x 64 channels.
// ---------------------------------------------------------------------------
__device__ __forceinline__ float tap_gather(const float* __restrict__ fb,
                                            float xi, float yi, int H, int W) {
  const bool valid = (xi >= 0.0f) && (xi < (float)W) && (yi >= 0.0f) && (yi < (float)H);
  const int xc = (int)fminf(fmaxf(xi, 0.0f), (float)(W - 1));
  const int yc = (int)fminf(fmaxf(yi, 0.0f), (float)(H - 1));
  return valid ? fb[yc * W + xc] : 0.0f;
}

__global__ __launch_bounds__(256) void sample_kernel(
    const float* __restrict__ offs,   // [600,384]
    const float* __restrict__ xyzr,   // [600,4]
    const float* __restrict__ f0, const float* __restrict__ f1,
    const float* __restrict__ f2, const float* __restrict__ f3,
    float* __restrict__ sampled) {    // [600,4,32,64]
  const int c  = threadIdx.x & 63;
  const int pt = blockIdx.x * 4 + (threadIdx.x >> 6);   // < 76800
  const int p  = pt & 31;
  const int g  = (pt >> 5) & 3;
  const int bn = pt >> 7;
  const int b  = bn / 300;

  const float x = xyzr[bn * 4 + 0];
  const float y = xyzr[bn * 4 + 1];
  const float z = xyzr[bn * 4 + 2];
  const float r = xyzr[bn * 4 + 3];
  const float* ob = offs + (size_t)bn * 384 + (size_t)(g * 32 + p) * 3;
  const float roi_w = exp2f(z - 0.5f * r);
  const float roi_h = exp2f(z + 0.5f * r);
  const float sx = x + ob[0] * roi_w;
  const float sy = y + ob[1] * roi_h;
  const float sz = z + ob[2];

  float wl[4], wsum = 0.0f;
  for (int l = 0; l < 4; ++l) {
    const float dz = sz - (float)(2 + l);     // log2 strides = 2,3,4,5
    wl[l] = __expf(-dz * dz * 0.5f);          // TAU = 2
    wsum += wl[l];
  }
  const float winv = 1.0f / wsum;

  const float* fs[4] = {f0, f1, f2, f3};
  const int Hs[4] = {200, 100, 50, 25};
  const int Ws[4] = {336, 168, 84, 42};

  float acc = 0.0f;
  for (int l = 0; l < 4; ++l) {
    const float stride = (float)(4 << l);
    const float xp = sx / stride - 0.5f;
    const float yp = sy / stride - 0.5f;
    const float x0 = floorf(xp), y0 = floorf(yp);
    const float wx1 = xp - x0, wy1 = yp - y0;
    const float wx0 = 1.0f - wx1, wy0 = 1.0f - wy1;
    const int H = Hs[l], W = Ws[l];
    const float* fb = fs[l] + (size_t)((b * 256) + g * 64 + c) * H * W;
    const float v00 = tap_gather(fb, x0,        y0,        H, W);
    const float v10 = tap_gather(fb, x0 + 1.0f, y0,        H, W);
    const float v01 = tap_gather(fb, x0,        y0 + 1.0f, H, W);
    const float v11 = tap_gather(fb, x0 + 1.0f, y0 + 1.0f, H, W);
    acc += (wl[l] * winv) *
           (v00 * (wx0 * wy0) + v10 * (wx1 * wy0) +
            v01 * (wx0 * wy1) + v11 * (wx1 * wy1));
  }
  sampled[(size_t)pt * 64 + c] = acc;
}

// ---------------------------------------------------------------------------
// Per-(b,n,g) adaptive mixing, LDS-resident, WMMA matmuls:
//   h1 = relu(LN2(sampled(32x64) @ M(64x64)))
//   h2 = relu(LN2(S(128x32) @ h1(32x64)))  -> hflat
// ---------------------------------------------------------------------------
#define SMP_P 68   // padded pitch for A-operand reads (bank conflict free)
#define SS_P  36

__global__ __launch_bounds__(256) void mix_kernel(
    const float* __restrict__ sampled,  // [gid][32][64]
    const float* __restrict__ params,   // [bn][32768]; per-g: M(4096)+S(4096)
    float* __restrict__ hflat) {        // [bn][32768] = (g, q(128), d(64))
  __shared__ float s_samp[32 * SMP_P];
  __shared__ float s_M[64 * 64];
  __shared__ float s_S[128 * SS_P];
  __shared__ float s_h1[32 * 64];
  __shared__ float s_red[4];

  const int tid = threadIdx.x;
  const int gid = blockIdx.x;           // (b*300+n)*4 + g
  const int g   = gid & 3;
  const int bn  = gid >> 2;
  const float* sg = sampled + (size_t)gid * 2048;
  const float* pg = params + (size_t)bn * 32768 + (size_t)g * 8192;

#pragma unroll
  for (int j = 0; j < 2; ++j) {                 // sampled: 512 float4
    const int e = tid + j * 256;
    const int row = e >> 4, c4 = e & 15;
    *(float4*)&s_samp[row * SMP_P + c4 * 4] = ((const float4*)sg)[e];
  }
#pragma unroll
  for (int j = 0; j < 4; ++j) {                 // M: 1024 float4
    const int e = tid + j * 256;
    ((float4*)s_M)[e] = ((const float4*)pg)[e];
  }
#pragma unroll
  for (int j = 0; j < 4; ++j) {                 // S: 1024 float4
    const int e = tid + j * 256;
    const int row = e >> 3, c4 = e & 7;
    *(float4*)&s_S[row * SS_P + c4 * 4] = ((const float4*)(pg + 4096))[e];
  }
  if (tid < 4) s_red[tid] = 0.0f;
  __syncthreads();

  const int lane = tid & 31, wave = tid >> 5;
  const int half = lane >> 4, lr = lane & 15;

  // ---- stage 1: 32x64 = 32x64 @ 64x64 ; one 16x16 tile per wave ----
  const int tm = wave >> 2, tn = wave & 3;
  v8f acc = {};
#pragma unroll
  for (int k = 0; k < 64; k += 4) {
    const int ka = k + 2 * half;
    v2f a = *(const v2f*)&s_samp[(tm * 16 + lr) * SMP_P + ka];
    v2f b;
    b.x = s_M[ka * 64 + tn * 16 + lr];
    b.y = s_M[(ka + 1) * 64 + tn * 16 + lr];
    acc = wmma4(a, b, acc);
  }
  float sm = 0.0f, sq = 0.0f;
#pragma unroll
  for (int r = 0; r < 8; ++r) { const float v = acc[r]; sm += v; sq += v * v; }
  atomicAdd(&s_red[0], sm);
  atomicAdd(&s_red[1], sq);
  __syncthreads();
  {
    const float mu  = s_red[0] * (1.0f / 2048.0f);
    const float var = s_red[1] * (1.0f / 2048.0f) - mu * mu;
    const float rs  = rsqrtf(var + 1e-5f);
#pragma unroll
    for (int r = 0; r < 8; ++r) {
      const float v = fmaxf((acc[r] - mu) * rs, 0.0f);
      s_h1[(tm * 16 + r + 8 * half) * 64 + tn * 16 + lr] = v;
    }
  }
  __syncthreads();

  // ---- stage 2: 128x64 = 128x32 @ 32x64 ; wave w owns rows w*16..w*16+15 ----
  v8f acc2[4] = {};
#pragma unroll
  for (int k = 0; k < 32; k += 4) {
    const int ka = k + 2 * half;
    v2f a = *(const v2f*)&s_S[(wave * 16 + lr) * SS_P + ka];
#pragma unroll
    for (int i = 0; i < 4; ++i) {
      v2f b;
      b.x = s_h1[ka * 64 + i * 16 + lr];
      b.y = s_h1[(ka + 1) * 64 + i * 16 + lr];
      acc2[i] = wmma4(a, b, acc2[i]);
    }
  }
  float sm2 = 0.0f, sq2 = 0.0f;
#pragma unroll
  for (int i = 0; i < 4; ++i)
#pragma unroll
    for (int r = 0; r < 8; ++r) { const float v = acc2[i][r]; sm2 += v; sq2 += v * v; }
  atomicAdd(&s_red[2], sm2);
  atomicAdd(&s_red[3], sq2);
  __syncthreads();
  const float mu2  = s_red[2] * (1.0f / 8192.0f);
  const float var2 = s_red[3] * (1.0f / 8192.0f) - mu2 * mu2;
  const float rs2  = rsqrtf(var2 + 1e-5f);
  float* outp = hflat + (size_t)bn * 32768 + (size_t)g * 8192;
#pragma unroll
  for (int i = 0; i < 4; ++i)
#pragma unroll
    for (int r = 0; r < 8; ++r) {
      const float v = fmaxf((acc2[i][r] - mu2) * rs2, 0.0f);
      const int q = wave * 16 + r + 8 * half;
      const int d = i * 16 + lr;
      outp[q * 64 + d] = v;
    }
}

// ---------------------------------------------------------------------------
// Split-K reduce + residual + bias + row LayerNorm -> out[600,256]
// ---------------------------------------------------------------------------
__global__ __launch_bounds__(256) void reduce_ln_kernel(
    const float* __restrict__ part,     // [8][600][256]
    const float* __restrict__ queries, const float* __restrict__ b_out,
    const float* __restrict__ gamma, const float* __restrict__ beta,
    float* __restrict__ out) {
  __shared__ float s_red[2];
  const int row = blockIdx.x, tid = threadIdx.x;
  float x = queries[(size_t)row * 256 + tid] + b_out[tid];
#pragma unroll
  for (int s = 0; s < 8; ++s)
    x += part[((size_t)s * 600 + row) * 256 + tid];
  if (tid < 2) s_red[tid] = 0.0f;
  __syncthreads();
  atomicAdd(&s_red[0], x);
  atomicAdd(&s_red[1], x * x);
  __syncthreads();
  const float mu  = s_red[0] * (1.0f / 256.0f);
  const float var = s_red[1] * (1.0f / 256.0f) - mu * mu;
  out[(size_t)row * 256 + tid] =
      (x - mu) * rsqrtf(var + 1e-5f) * gamma[tid] + beta[tid];
}

// ---------------------------------------------------------------------------
// Launch
// ---------------------------------------------------------------------------
extern "C" void kernel_launch(void* const* d_in, const int* in_sizes, int n_in,
                              void* d_out, int out_size, void* d_ws, size_t ws_size,
                              hipStream_t stream) {
  (void)in_sizes; (void)n_in; (void)out_size;
  const float* queries = (const float*)d_in[0];   // [2,300,256]
  const float* xyzr    = (const float*)d_in[1];   // [2,300,4]
  const float* f0      = (const float*)d_in[2];
  const float* f1      = (const float*)d_in[3];
  const float* f2      = (const float*)d_in[4];
  const float* f3      = (const float*)d_in[5];
  const float* W_off   = (const float*)d_in[6];   // [256,384]
  const float* b_off   = (const float*)d_in[7];   // [384]
  const float* W_param = (const float*)d_in[8];   // [256,32768]
  const float* b_param = (const float*)d_in[9];   // [32768]
  const float* W_out   = (const float*)d_in[10];  // [32768,256]
  const float* b_out   = (const float*)d_in[11];  // [256]
  const float* gamma   = (const float*)d_in[12];  // [256]
  const float* beta    = (const float*)d_in[13];  // [256]
  float* out = (float*)d_out;                     // [600,256]

  const size_t nOffs = 600u * 384u;
  const size_t nSamp = 600u * 4u * 32u * 64u;
  const size_t nPar  = 600u * 32768u;
  const size_t nPart = 8u * 600u * 256u;
  const size_t needed = (nOffs + nSamp + 2 * nPar + nPart) * sizeof(float);
  if (ws_size < needed) return;

  float* w_offs = (float*)d_ws;
  float* w_samp = w_offs + nOffs;
  float* w_par  = w_samp + nSamp;
  float* w_h    = w_par + nPar;
  float* w_part = w_h + nPar;

  const int ntM = (600 + MBLK - 1) / MBLK;  // 19

  // 1) offsets = Q @ W_off + b_off            (600x384, K=256)
  gemm16_kernel<<<dim3(ntM * (384 / NBLK), 1), 256, 0, stream>>>(
      queries, 256, W_off, b_off, nullptr, w_offs, 600, 384, 256, 0, 0);
  // 2) params = Q @ W_param + b_param         (600x32768, K=256)
  gemm16_kernel<<<dim3(ntM * (32768 / NBLK), 1), 256, 0, stream>>>(
      queries, 256, W_param, b_param, nullptr, w_par, 600, 32768, 256, 0, 0);
  // 3) adaptive sampling -> sampled
  sample_kernel<<<76800 / 4, 256, 0, stream>>>(
      w_offs, xyzr, f0, f1, f2, f3, w_samp);
  // 4) per-group mixing (2 WMMA matmuls + 2 LN) -> hflat
  mix_kernel<<<2 * 300 * 4, 256, 0, stream>>>(w_samp, w_par, w_h);
  // 5) split-K (8 x K=4096) partials: part[s] = hflat[:,s] @ W_out[s,:]
  gemm16_kernel<<<dim3(ntM * (256 / NBLK), 8), 256, 0, stream>>>(
      w_h, 32768, W_out, nullptr, nullptr, w_part, 600, 256, 4096,
      /*kStep=*/4096, /*cStride=*/(unsigned long long)600 * 256);
  // 6) reduce partials + residual + bias + row LayerNorm
  reduce_ln_kernel<<<600, 256, 0, stream>>>(
      w_part, queries, b_out, gamma, beta, out);
}